// DifferentialGraphTransformerLayer_8615704396326
// MI455X (gfx1250) — compile-verified
//
#include <hip/hip_runtime.h>
#include <math.h>

// Problem constants (match reference)
#define Nn   50000
#define Ee   800000
#define E2   (Nn + Ee)
#define HH   8
#define CC   16
#define HCv  128
#define FFv  512
#define DIN  128
#define NEGs 0.2f
#define EPSs 1e-5f

typedef float v2f __attribute__((ext_vector_type(2)));
typedef float v8f __attribute__((ext_vector_type(8)));

// ---------- helpers ----------
__device__ __forceinline__ unsigned fenc(float f) {
    unsigned u = __float_as_uint(f);
    return (u & 0x80000000u) ? ~u : (u | 0x80000000u); // monotone order-preserving
}
__device__ __forceinline__ float fdec(unsigned u) {
    return (u & 0x80000000u) ? __uint_as_float(u ^ 0x80000000u) : __uint_as_float(~u);
}
__device__ __forceinline__ float gelu_exact(float x) {
    return 0.5f * x * (1.0f + erff(x * 0.70710678118654752f));
}

// ---------- pack B [K,N] row-major into WMMA fragment order ----------
// Bp[((kblk*(N/16) + jtile)*32 + lane)*2 + {0,1}] = B[kblk*4 + 2*half + {0,1}][jtile*16 + r]
// so each lane's B fragment for (kblk, jtile) is one contiguous 8-byte pair.
__global__ void pack_b_kernel(const float* __restrict__ B, float* __restrict__ Bp, int N, int K)
{
    int t = blockIdx.x * blockDim.x + threadIdx.x;   // one thread per fragment-lane
    int totalLanes = (K >> 2) * (N >> 4) * 32;
    if (t >= totalLanes) return;
    int lane  = t & 31;
    int rest  = t >> 5;                 // kblk*(N/16) + jtile
    int half  = lane >> 4, r = lane & 15;
    int jt    = rest % (N >> 4);
    int kblk  = rest / (N >> 4);
    int k     = (kblk << 2) + 2 * half;
    int col   = (jt << 4) + r;
    Bp[(size_t)t * 2 + 0] = B[(size_t)k * N + col];
    Bp[(size_t)t * 2 + 1] = B[(size_t)(k + 1) * N + col];
}

// ---------- fp32 WMMA GEMM, 16x128 strip per wave, pre-packed B ----------
// C = act(A@B + bias) + residual.  A: [M,K] row-major, Bp: packed fragments.
// M%16==0, N%128==0, K%4==0. Per K-step: 1 b64 A load + 8 b64 B loads + 8 WMMAs.
__global__ void wmma_gemm_f32_strip(const float* __restrict__ A, const float* __restrict__ Bp,
                                    const float* __restrict__ bias, const float* __restrict__ res,
                                    float* __restrict__ C, int M, int N, int K, int act)
{
    const int lane   = threadIdx.x & 31;
    const int wpb    = blockDim.x >> 5;
    const int wave   = blockIdx.x * wpb + (threadIdx.x >> 5);
    const int strips = N >> 7;                       // 128-wide strips
    const int total  = (M >> 4) * strips;
    if (wave >= total) return;                       // wave-uniform: EXEC stays all-ones
    const int m0   = (wave / strips) << 4;
    const int n0   = (wave % strips) << 7;
    const int half = lane >> 4;                      // 0 -> K{0,1}, 1 -> K{2,3}
    const int r    = lane & 15;

    v8f acc[8];
#pragma unroll
    for (int j = 0; j < 8; ++j) acc[j] = (v8f){0.f,0.f,0.f,0.f,0.f,0.f,0.f,0.f};

    const float* arow = A + (size_t)(m0 + r) * K + 2 * half;       // 8B-aligned pairs
    const float* bbase = Bp + ((size_t)(n0 >> 4) * 64) + lane * 2; // jtile base within k-quad
    const int kqStride = (N >> 4) << 6;                            // floats per k-quad row

    for (int k0 = 0; k0 < K; k0 += 4) {
        v2f a = *(const v2f*)(arow + k0);
        const float* bq = bbase + (size_t)(k0 >> 2) * kqStride;
        v2f bf[8];
#pragma unroll
        for (int j = 0; j < 8; ++j)
            bf[j] = *(const v2f*)(bq + (size_t)j * 64);
#pragma unroll
        for (int j = 0; j < 8; ++j)
            acc[j] = __builtin_amdgcn_wmma_f32_16x16x4_f32(false, a, false, bf[j],
                                                           (short)0, acc[j], false, false);
    }

#pragma unroll
    for (int j = 0; j < 8; ++j) {
        const int col = n0 + j * 16 + r;
        const float bv = bias ? bias[col] : 0.0f;
#pragma unroll
        for (int i = 0; i < 8; ++i) {
            const int row = m0 + i + 8 * half;       // C layout: VGPR i -> M = i + 8*half
            float v = acc[j][i] + bv;
            if (act) v = gelu_exact(v);
            if (res) v += res[(size_t)row * N + col];
            C[(size_t)row * N + col] = v;
        }
    }
}

// ---------- self-loop attribute accumulation ----------
__global__ void deg_loop_kernel(const float* __restrict__ ea, const int* __restrict__ ei,
                                float* __restrict__ deg, float* __restrict__ loopat)
{
    int t = blockIdx.x * blockDim.x + threadIdx.x;
    if (t >= Ee * CC) return;
    int edge = t >> 4, c = t & 15;
    int d = ei[Ee + edge];                       // edge_index[1] = dst
    atomicAdd(&loopat[(size_t)d * CC + c], ea[(size_t)edge * CC + c]);
    if (c == 0) atomicAdd(&deg[d], 1.0f);
}

__global__ void loop_div_kernel(const float* __restrict__ deg, float* __restrict__ loopat)
{
    int t = blockIdx.x * blockDim.x + threadIdx.x;
    if (t >= Nn * CC) return;
    loopat[t] /= fmaxf(deg[t >> 4], 1.0f);
}

// ---------- tiny precompute: lambda per head, combined output bias ----------
__global__ void small_precompute(const float* __restrict__ lq1, const float* __restrict__ lk1,
                                 const float* __restrict__ lq2, const float* __restrict__ lk2,
                                 const float* __restrict__ b_out, const float* __restrict__ b_ex,
                                 float* __restrict__ lam, float* __restrict__ bias_comb)
{
    int i = threadIdx.x;
    if (i < HH) {
        float s1 = 0.f, s2 = 0.f;
        for (int c = 0; c < CC; ++c) {
            s1 += lq1[i * CC + c] * lk1[i * CC + c];
            s2 += lq2[i * CC + c] * lk2[i * CC + c];
        }
        lam[i] = expf(s1) - expf(s2) + 0.8f;   // LAMBDA_INIT
    }
    if (i < HCv) bias_comb[i] = b_out[i] + b_ex[i];
}

// ---------- attention pass kernels: one thread per (edge, head) ----------
__global__ void alpha_max_kernel(const float* __restrict__ xh, const float* __restrict__ e,
                                 const float* __restrict__ att_l, const float* __restrict__ att_r,
                                 const float* __restrict__ att_e, const int* __restrict__ ei,
                                 float* __restrict__ alpha, unsigned* __restrict__ amax)
{
    int t = blockIdx.x * blockDim.x + threadIdx.x;
    if (t >= E2 * HH) return;
    int edge = t >> 3, hh = t & 7;
    int s, d;
    if (edge < Ee) { s = ei[edge]; d = ei[Ee + edge]; } else { s = d = edge - Ee; }
    const float* xi = xh + (size_t)d * HCv + hh * CC;
    const float* xj = xh + (size_t)s * HCv + hh * CC;
    const float* ee = e  + (size_t)edge * HCv + hh * CC;
    const float* al = att_l + hh * CC;
    const float* ar = att_r + hh * CC;
    const float* ae = att_e + hh * CC;
    float a = 0.f;
#pragma unroll
    for (int c = 0; c < CC; ++c)
        a += xi[c] * al[c] + xj[c] * ar[c] + ee[c] * ae[c];
    a = (a >= 0.f) ? a : NEGs * a;              // leaky_relu
    alpha[t] = a;
    atomicMax(&amax[(size_t)d * HH + hh], fenc(a));
}

__global__ void alpha_exp_kernel(float* __restrict__ alpha, const unsigned* __restrict__ amax,
                                 float* __restrict__ denom, const int* __restrict__ ei)
{
    int t = blockIdx.x * blockDim.x + threadIdx.x;
    if (t >= E2 * HH) return;
    int edge = t >> 3, hh = t & 7;
    int d = (edge < Ee) ? ei[Ee + edge] : (edge - Ee);
    float a = expf(alpha[t] - fdec(amax[(size_t)d * HH + hh]));
    alpha[t] = a;
    atomicAdd(&denom[(size_t)d * HH + hh], a);
}

__global__ void scatter_kernel(const float* __restrict__ xh, const float* __restrict__ e,
                               const float* __restrict__ alpha, const float* __restrict__ denom,
                               const int* __restrict__ ei, float* __restrict__ out)
{
    int t = blockIdx.x * blockDim.x + threadIdx.x;
    if (t >= E2 * HH) return;
    int edge = t >> 3, hh = t & 7;
    int s, d;
    if (edge < Ee) { s = ei[edge]; d = ei[Ee + edge]; } else { s = d = edge - Ee; }
    float w = alpha[t] / denom[(size_t)d * HH + hh];
    const float* xj = xh + (size_t)s * HCv + hh * CC;
    const float* ee = e  + (size_t)edge * HCv + hh * CC;
    float* o = out + (size_t)d * HCv + hh * CC;
#pragma unroll
    for (int c = 0; c < CC; ++c)
        atomicAdd(&o[c], (xj[c] + ee[c]) * w);
}

// ---------- differential combine + RMSNorm*(1-lambda_init) ----------
__global__ void combine_rms_kernel(const float* __restrict__ o1, const float* __restrict__ o2,
                                   const float* __restrict__ lam, const float* __restrict__ w,
                                   float* __restrict__ out)
{
    int n = blockIdx.x, i = threadIdx.x;
    float v = o1[(size_t)n * HCv + i] - lam[i >> 4] * o2[(size_t)n * HCv + i];
    __shared__ float red[HCv];
    red[i] = v * v;
    __syncthreads();
    for (int s = HCv / 2; s > 0; s >>= 1) {
        if (i < s) red[i] += red[i + s];
        __syncthreads();
    }
    float norm = sqrtf(red[0] / (float)HCv + EPSs);
    out[(size_t)n * HCv + i] = v / norm * w[i] * 0.2f;   // *(1 - LAMBDA_INIT)
}

// ---------- plain RMSNorm ----------
__global__ void rms_kernel(const float* __restrict__ in, const float* __restrict__ w,
                           float* __restrict__ out)
{
    int n = blockIdx.x, i = threadIdx.x;
    float v = in[(size_t)n * HCv + i];
    __shared__ float red[HCv];
    red[i] = v * v;
    __syncthreads();
    for (int s = HCv / 2; s > 0; s >>= 1) {
        if (i < s) red[i] += red[i + s];
        __syncthreads();
    }
    float norm = sqrtf(red[0] / (float)HCv + EPSs);
    out[(size_t)n * HCv + i] = v / norm * w[i];
}

// ---------- host launch ----------
extern "C" void kernel_launch(void* const* d_in, const int* in_sizes, int n_in,
                              void* d_out, int out_size, void* d_ws, size_t ws_size,
                              hipStream_t stream)
{
    (void)in_sizes; (void)n_in; (void)out_size; (void)ws_size;

    const float* x         = (const float*)d_in[0];
    const float* edge_attr = (const float*)d_in[1];
    const float* W_l1      = (const float*)d_in[2];
    const float* W_l2      = (const float*)d_in[3];
    const float* att_l1    = (const float*)d_in[4];
    const float* att_r1    = (const float*)d_in[5];
    const float* att_l2    = (const float*)d_in[6];
    const float* att_r2    = (const float*)d_in[7];
    const float* W_e1      = (const float*)d_in[8];
    const float* att_e1    = (const float*)d_in[9];
    const float* W_e2      = (const float*)d_in[10];
    const float* att_e2    = (const float*)d_in[11];
    const float* lq1       = (const float*)d_in[12];
    const float* lk1       = (const float*)d_in[13];
    const float* lq2       = (const float*)d_in[14];
    const float* lk2       = (const float*)d_in[15];
    const float* rms_attn  = (const float*)d_in[16];
    const float* W_out     = (const float*)d_in[17];
    const float* b_out     = (const float*)d_in[18];
    const float* b_extra   = (const float*)d_in[19];
    const float* rms_layer = (const float*)d_in[20];
    const float* W_ff1     = (const float*)d_in[21];
    const float* b_ff1     = (const float*)d_in[22];
    const float* W_ff2     = (const float*)d_in[23];
    const float* b_ff2     = (const float*)d_in[24];
    const int*   ei        = (const int*)d_in[25];

    // Workspace layout (floats)
    float*    ws      = (float*)d_ws;
    float*    deg     = ws;                                    // Nn
    float*    loopat  = deg    + (size_t)Nn;                   // Nn*16
    float*    denom   = loopat + (size_t)Nn * CC;              // Nn*8
    unsigned* amax    = (unsigned*)(denom + (size_t)Nn * HH);  // Nn*8
    float*    out1    = (float*)(amax + (size_t)Nn * HH);      // Nn*128
    float*    out2    = out1   + (size_t)Nn * HCv;             // Nn*128
    const size_t zeroElems = (size_t)Nn * (1 + CC + HH + HH + HCv + HCv);
    float*    x1      = out2   + (size_t)Nn * HCv;             // Nn*128
    float*    x2      = x1     + (size_t)Nn * HCv;             // Nn*128
    float*    lam     = x2     + (size_t)Nn * HCv;             // 16 (8 used)
    float*    bcomb   = lam    + 16;                           // 128
    float*    attn    = bcomb  + HCv;                          // Nn*128
    float*    h       = attn   + (size_t)Nn * HCv;             // Nn*128
    float*    hn      = h      + (size_t)Nn * HCv;             // Nn*128
    float*    ff1     = hn     + (size_t)Nn * HCv;             // Nn*512
    float*    alpha   = ff1    + (size_t)Nn * FFv;             // E2*8
    float*    bpack   = alpha  + (size_t)E2 * HH;              // up to 512*128 = 64K floats
    float*    ebuf    = bpack  + (size_t)FFv * HCv;            // E2*128 (reused per pass)

    // zero: deg, loop_attr, denom, amax (0 == encoded -inf), out1, out2
    hipMemsetAsync(deg, 0, zeroElems * sizeof(float), stream);

    auto packB = [&](const float* B, int N, int K) {
        int totalLanes = (K >> 2) * (N >> 4) * 32;
        pack_b_kernel<<<(totalLanes + 255) / 256, 256, 0, stream>>>(B, bpack, N, K);
    };
    auto gemm = [&](const float* A, const float* bias, const float* res,
                    float* C, int M, int N, int K, int act) {
        int waves = (M >> 4) * (N >> 7);             // one wave per 16x128 strip
        int grid  = (waves + 3) / 4;                 // 4 waves per 128-thread block
        wmma_gemm_f32_strip<<<grid, 128, 0, stream>>>(A, bpack, bias, res, C, M, N, K, act);
    };

    // 1) self-loop edge attributes (mean of incoming)
    deg_loop_kernel<<<(Ee * CC + 255) / 256, 256, 0, stream>>>(edge_attr, ei, deg, loopat);
    loop_div_kernel<<<(Nn * CC + 255) / 256, 256, 0, stream>>>(deg, loopat);

    // 2) node projections
    packB(W_l1, HCv, DIN);
    gemm(x, nullptr, nullptr, x1, Nn, HCv, DIN, 0);
    packB(W_l2, HCv, DIN);
    gemm(x, nullptr, nullptr, x2, Nn, HCv, DIN, 0);

    // 3) lambda + combined bias
    small_precompute<<<1, 128, 0, stream>>>(lq1, lk1, lq2, lk2, b_out, b_extra, lam, bcomb);

    const int tEH = E2 * HH;

    // ---- attention pass 1 ----
    packB(W_e1, HCv, CC);
    gemm(edge_attr, nullptr, nullptr, ebuf, Ee, HCv, CC, 0);
    gemm(loopat,    nullptr, nullptr, ebuf + (size_t)Ee * HCv, Nn, HCv, CC, 0);
    alpha_max_kernel<<<(tEH + 255) / 256, 256, 0, stream>>>(x1, ebuf, att_l1, att_r1, att_e1, ei, alpha, amax);
    alpha_exp_kernel<<<(tEH + 255) / 256, 256, 0, stream>>>(alpha, amax, denom, ei);
    scatter_kernel  <<<(tEH + 255) / 256, 256, 0, stream>>>(x1, ebuf, alpha, denom, ei, out1);

    // reset denom + amax (contiguous) for pass 2
    hipMemsetAsync(denom, 0, (size_t)Nn * HH * 2 * sizeof(float), stream);

    // ---- attention pass 2 ----
    packB(W_e2, HCv, CC);
    gemm(edge_attr, nullptr, nullptr, ebuf, Ee, HCv, CC, 0);
    gemm(loopat,    nullptr, nullptr, ebuf + (size_t)Ee * HCv, Nn, HCv, CC, 0);
    alpha_max_kernel<<<(tEH + 255) / 256, 256, 0, stream>>>(x2, ebuf, att_l2, att_r2, att_e2, ei, alpha, amax);
    alpha_exp_kernel<<<(tEH + 255) / 256, 256, 0, stream>>>(alpha, amax, denom, ei);
    scatter_kernel  <<<(tEH + 255) / 256, 256, 0, stream>>>(x2, ebuf, alpha, denom, ei, out2);

    // 4) differential combine + RMSNorm, output projection with residual x
    combine_rms_kernel<<<Nn, 128, 0, stream>>>(out1, out2, lam, rms_attn, attn);
    packB(W_out, HCv, HCv);
    gemm(attn, bcomb, x, h, Nn, HCv, HCv, 0);

    // 5) pre-norm FFN with residual h
    rms_kernel<<<Nn, 128, 0, stream>>>(h, rms_layer, hn);
    packB(W_ff1, FFv, HCv);
    gemm(hn, b_ff1, nullptr, ff1, Nn, FFv, HCv, 1);              // exact GELU
    packB(W_ff2, HCv, FFv);
    gemm(ff1, b_ff2, h, (float*)d_out, Nn, HCv, FFv, 0);         // final = h + ffn
}